// HGCN_21311627722996
// MI455X (gfx1250) — compile-verified
//
#include <hip/hip_runtime.h>
#include <math.h>

// ---- reference constants ----
#define MIN_NORM  1e-15f
#define EPSF      1e-7f
#define MAX_NORMF 1000000.0f
#define LN_EPSF   1e-5f
// K = SQRT_K = 1, NORM_FACTOR = 100

#define KPAD 264   // 258 real K columns padded to multiple of 8
#define NKG  66    // KPAD/4 K-groups (one WMMA each)
#define QROW 268   // LDS row stride: 268%64=12 -> conflict-free A-frag reads

typedef __attribute__((ext_vector_type(2))) float v2f;
typedef __attribute__((ext_vector_type(8))) float v8f;

__device__ __forceinline__ float wsum32(float v) {
#pragma unroll
  for (int m = 16; m > 0; m >>= 1) v += __shfl_xor(v, m, 32);
  return v;
}
// reduce across a 16-lane half-wave (masks < 16 stay inside the half)
__device__ __forceinline__ float hsum16(float v) {
#pragma unroll
  for (int m = 8; m > 0; m >>= 1) v += __shfl_xor(v, m, 32);
  return v;
}
__device__ __forceinline__ float arcosh_f(float t) {
  return logf(t + sqrtf(fmaxf(t * t - 1.f, 0.f)));
}
__device__ __forceinline__ float siluf(float z) { return z / (1.f + expf(-z)); }

// =====================================================================
// Kernel 0: repack aW1 (128 x 258) into per-lane WMMA B-fragment order:
// Bpack[((n*NKG + kg)*32 + lane)] = v2f{ B[kg*4+koff][o], B[kg*4+koff+1][o] }
// with o = n*16 + (lane&15), koff = (lane>=16)*2; zero-padded K >= 258.
// One global_load_b64 per fragment in the GEMM loop, fully coalesced.
// =====================================================================
__global__ void pack_aW1_kernel(const float* __restrict__ aW1,
                                float* __restrict__ Bpack) {
  int t = blockIdx.x * blockDim.x + threadIdx.x;  // over 8 * NKG * 32
  if (t >= 8 * NKG * 32) return;
  int lane = t & 31;
  int kg   = (t >> 5) % NKG;
  int n    = t / (NKG * 32);
  int m    = lane & 15;
  int koff = (lane >> 4) * 2;
  int o    = n * 16 + m;
  int k    = kg * 4 + koff;
  Bpack[2 * t]     = (k     < 258) ? aW1[(size_t)o * 258 + k]     : 0.f;
  Bpack[2 * t + 1] = (k + 1 < 258) ? aW1[(size_t)o * 258 + k + 1] : 0.f;
}

// =====================================================================
// Kernel 1: per-node  x = mobius_add(expmap0(proj_tan0(logmap0(h) @ W^T)),
//                                    expmap0(proj_tan0(bias)))
//           xt = logmap0(x)
// One wave per node; lane holds components lane*4 .. lane*4+3.
// blockDim = 128 (4 waves), grid = N/4.
// =====================================================================
__global__ void node_linear_kernel(const float* h, int Din,
                                   const float* __restrict__ W,
                                   const float* __restrict__ bias,
                                   float* x_out, float* xt_out) {
  __shared__ float u_lds[4][128];
  const int lane = threadIdx.x & 31;
  const int wid  = threadIdx.x >> 5;
  const int node = blockIdx.x * 4 + wid;
  const float* hr = h + (size_t)node * Din;

  // ---- u = logmap0(h row) staged to LDS ----
  float part = 0.f;
  for (int c = lane; c < Din; c += 32) { float v = hr[c]; if (c > 0) part += v * v; }
  float yn = fmaxf(sqrtf(wsum32(part)), MIN_NORM);
  float sc = arcosh_f(fmaxf(hr[0], 1.f + EPSF)) / yn;
  for (int c = lane; c < 128; c += 32)
    u_lds[wid][c] = (c == 0 || c >= Din) ? 0.f : sc * hr[c];
  __syncthreads();

  // ---- x = u @ W^T (4 outputs per lane) ----
  float xv[4];
#pragma unroll
  for (int j = 0; j < 4; j++) {
    int o = lane * 4 + j;
    float acc = 0.f;
    for (int k = 0; k < Din; k++) acc += u_lds[wid][k] * W[(size_t)o * Din + k];
    xv[j] = acc;
  }
  // ---- proj_tan0 + expmap0 ----
  part = 0.f;
#pragma unroll
  for (int j = 0; j < 4; j++) { int c = lane * 4 + j; if (c > 0) part += xv[j] * xv[j]; }
  float xn = fmaxf(sqrtf(wsum32(part)), MIN_NORM);
  float ch = coshf(xn), sh = sinhf(xn) / xn;
#pragma unroll
  for (int j = 0; j < 4; j++) { int c = lane * 4 + j; xv[j] = (c == 0) ? ch : sh * xv[j]; }

  // ---- mobius_add(x, expmap0(proj_tan0(bias))) ----
  // logmap0(expmap0(proj_tan0(b))) == proj_tan0(b)  (exact for these ranges)
  float ub[4];
#pragma unroll
  for (int j = 0; j < 4; j++) { int c = lane * 4 + j; ub[j] = (c == 0) ? 0.f : bias[c]; }
  float x0 = __shfl(xv[0], 0, 32);
  part = 0.f;
#pragma unroll
  for (int j = 0; j < 4; j++) { int c = lane * 4 + j; if (c > 0) part += xv[j] * xv[j]; }
  float ynx = fmaxf(sqrtf(wsum32(part)), MIN_NORM);
  part = 0.f;
#pragma unroll
  for (int j = 0; j < 4; j++) { int c = lane * 4 + j; if (c > 0) part += (xv[j] / ynx) * ub[j]; }
  float alpha = wsum32(part);
  float w[4];
#pragma unroll
  for (int j = 0; j < 4; j++) {
    int c = lane * 4 + j;
    w[j] = (c == 0) ? 0.f : ub[j] - alpha * (1.f - x0) * xv[j] / ynx;
  }
  part = 0.f;
#pragma unroll
  for (int j = 0; j < 4; j++) { int c = lane * 4 + j; if (c > 0) part += xv[j] * w[j]; }
  float t0 = wsum32(part) / fmaxf(x0, EPSF);
  // expmap(t, x), t = [t0, w_spatial]
  part = 0.f;
#pragma unroll
  for (int j = 0; j < 4; j++) { int c = lane * 4 + j; if (c > 0) part += w[j] * w[j]; }
  float md = wsum32(part) - t0 * t0;
  float nu = fminf(sqrtf(fmaxf(md, MIN_NORM)), MAX_NORMF);
  float theta = fmaxf(nu, MIN_NORM);
  float che = coshf(theta), shk = sinhf(theta) / theta;
#pragma unroll
  for (int j = 0; j < 4; j++) { int c = lane * 4 + j; if (c > 0) xv[j] = che * xv[j] + shk * w[j]; }
  part = 0.f;
#pragma unroll
  for (int j = 0; j < 4; j++) { int c = lane * 4 + j; if (c > 0) part += xv[j] * xv[j]; }
  float s2  = wsum32(part);
  float nx0 = sqrtf(1.f + s2);               // hproj
  if (lane == 0) xv[0] = nx0;

  // ---- xt = logmap0(x) ----
  float ynn = fmaxf(sqrtf(s2), MIN_NORM);
  float sc2 = arcosh_f(fmaxf(nx0, 1.f + EPSF)) / ynn;
  float* xo  = x_out  + (size_t)node * 128;
  float* xto = xt_out + (size_t)node * 128;
#pragma unroll
  for (int j = 0; j < 4; j++) {
    int c = lane * 4 + j;
    xo[c]  = xv[j];
    xto[c] = (c == 0) ? 0.f : sc2 * xv[j];
  }
}

// =====================================================================
// Kernel 2: fused edge attention + aggregation (the WMMA kernel).
// One wave per 32-edge row-group (all share the same destination row r),
// processed as two 16-edge WMMA M-tiles. blockDim = 64 (2 waves).
// GEMM: n blocked by 2 (two live accumulators), K-loop kept ROLLED
// (#pragma unroll 1) so A fragments stay in LDS (no hoist->spill).
// Each K-iter: 1 shared A load (ds_load_2addr_b64), 4 coalesced B
// global_load_b64 (prefetched one iteration ahead), 4 WMMAs alternating
// accumulators (independent -> no WMMA->WMMA RAW stalls).
// =====================================================================
__global__ void edge_attn_kernel(const float* __restrict__ x,
                                 const float* __restrict__ xt,
                                 const float* __restrict__ dist,
                                 const float* __restrict__ emask,
                                 const float* __restrict__ Bpack,
                                 const float* __restrict__ ab1,
                                 const float* __restrict__ aW2,
                                 const float* __restrict__ ab2,
                                 const int* __restrict__ row,
                                 const int* __restrict__ col,
                                 float* __restrict__ out_agg) {
  __shared__ float qt[2][16][QROW];
  __shared__ float attL[2][16];
  __shared__ float mdL[2][16];
  __shared__ float geoL[2][16];

  const int lane = threadIdx.x & 31;
  const int wid  = threadIdx.x >> 5;
  const int grp  = blockIdx.x * 2 + wid;   // 32-edge group
  const int base = grp * 32;
  const int r    = row[base];              // same for all 32 edges

  float xr[4];
#pragma unroll
  for (int j = 0; j < 4; j++) xr[j] = x[(size_t)r * 128 + lane * 4 + j];
  const float xr0 = __shfl(xr[0], 0, 32);
  float aggv[4] = {0.f, 0.f, 0.f, 0.f};

  const int m    = lane & 15;        // WMMA M index within tile
  const int koff = (lane >> 4) * 2;  // K sub-offset per A/B layout

  for (int half = 0; half < 2; ++half) {
    // ---- mink_dot + sqdist per edge ----
    for (int e = 0; e < 16; e++) {
      int ge = base + half * 16 + e;
      int c  = col[ge];
      const float* xc = x + (size_t)c * 128;
      float p = 0.f;
#pragma unroll
      for (int j = 0; j < 4; j++) p += xr[j] * xc[lane * 4 + j];
      float dot = wsum32(p);
      float mdv = dot - 2.f * xr0 * xc[0];
      float th  = fmaxf(-mdv, 1.f + EPSF);
      float ac  = arcosh_f(th);
      float geo = fminf(ac * ac, 50.f);
      if (lane == 0) { mdL[wid][e] = mdv; geoL[wid][e] = geo; }
    }
    __syncthreads();
    // ---- stage q tile: [xt[r](128) | xt[c](128) | dist | geo | 0...] ----
    for (int e = 0; e < 16; e++) {
      int ge = base + half * 16 + e;
      int c  = col[ge];
      for (int k = lane; k < KPAD; k += 32) {
        float v;
        if (k < 128)       v = xt[(size_t)r * 128 + k];
        else if (k < 256)  v = xt[(size_t)c * 128 + (k - 128)];
        else if (k == 256) v = dist[ge];
        else if (k == 257) v = geoL[wid][e];
        else               v = 0.f;
        qt[wid][e][k] = v;
      }
    }
    __syncthreads();
    // ---- GEMM + fused logit, n blocked by 2 ----
    float lsum[8];
#pragma unroll
    for (int v = 0; v < 8; v++) lsum[v] = 0.f;

    for (int nb = 0; nb < 8; nb += 2) {
      const v2f* Bp0 = (const v2f*)Bpack + ((size_t)(nb)     * NKG) * 32 + lane;
      const v2f* Bp1 = (const v2f*)Bpack + ((size_t)(nb + 1) * NKG) * 32 + lane;
      v8f acc0 = {}, acc1 = {};
      v2f f00 = Bp0[0], f01 = Bp0[32];
      v2f f10 = Bp1[0], f11 = Bp1[32];
      // main loop: prefetch next iteration's B frags before the WMMAs
#pragma unroll 1
      for (int kg = 0; kg < NKG - 2; kg += 2) {
        v2f p00 = Bp0[(size_t)(kg + 2) * 32];
        v2f p01 = Bp0[(size_t)(kg + 3) * 32];
        v2f p10 = Bp1[(size_t)(kg + 2) * 32];
        v2f p11 = Bp1[(size_t)(kg + 3) * 32];
        int k0 = kg * 4;
        v2f a0, a1;
        a0.x = qt[wid][m][k0 + koff];
        a0.y = qt[wid][m][k0 + koff + 1];
        a1.x = qt[wid][m][k0 + 4 + koff];
        a1.y = qt[wid][m][k0 + 5 + koff];
        acc0 = __builtin_amdgcn_wmma_f32_16x16x4_f32(
            false, a0, false, f00, (short)0, acc0, false, false);
        acc1 = __builtin_amdgcn_wmma_f32_16x16x4_f32(
            false, a0, false, f10, (short)0, acc1, false, false);
        acc0 = __builtin_amdgcn_wmma_f32_16x16x4_f32(
            false, a1, false, f01, (short)0, acc0, false, false);
        acc1 = __builtin_amdgcn_wmma_f32_16x16x4_f32(
            false, a1, false, f11, (short)0, acc1, false, false);
        f00 = p00; f01 = p01; f10 = p10; f11 = p11;
      }
      // epilogue kg = NKG-2 (no prefetch)
      {
        const int k0 = (NKG - 2) * 4;
        v2f a0, a1;
        a0.x = qt[wid][m][k0 + koff];
        a0.y = qt[wid][m][k0 + koff + 1];
        a1.x = qt[wid][m][k0 + 4 + koff];
        a1.y = qt[wid][m][k0 + 5 + koff];
        acc0 = __builtin_amdgcn_wmma_f32_16x16x4_f32(
            false, a0, false, f00, (short)0, acc0, false, false);
        acc1 = __builtin_amdgcn_wmma_f32_16x16x4_f32(
            false, a0, false, f10, (short)0, acc1, false, false);
        acc0 = __builtin_amdgcn_wmma_f32_16x16x4_f32(
            false, a1, false, f01, (short)0, acc0, false, false);
        acc1 = __builtin_amdgcn_wmma_f32_16x16x4_f32(
            false, a1, false, f11, (short)0, acc1, false, false);
      }
      // fold both 16-column tiles into the logit partials, recycle accs
      int o0 = nb * 16 + m;
      int o1 = (nb + 1) * 16 + m;
      float b1v0 = ab1[o0], w2v0 = aW2[o0];
      float b1v1 = ab1[o1], w2v1 = aW2[o1];
#pragma unroll
      for (int v = 0; v < 8; v++) {
        lsum[v] += siluf(acc0[v] + b1v0) * w2v0;
        lsum[v] += siluf(acc1[v] + b1v1) * w2v1;
      }
    }
    // C layout: lane<16 -> M=v ; lane>=16 -> M=v+8 ; reduce over 16 lanes
    float logit[8];
#pragma unroll
    for (int v = 0; v < 8; v++) logit[v] = hsum16(lsum[v]);
    if (m == 0) {
      int mb = (lane >> 4) * 8;
#pragma unroll
      for (int v = 0; v < 8; v++) {
        int eloc = mb + v;
        int ge   = base + half * 16 + eloc;
        float a  = logit[v] + ab2[0];
        attL[wid][eloc] = (1.f / (1.f + expf(-a))) * emask[ge];
      }
    }
    __syncthreads();
    // ---- agg += logmap(x_r, x_c) * att ----
    for (int e = 0; e < 16; e++) {
      int ge = base + half * 16 + e;
      int c  = col[ge];
      const float* xc = x + (size_t)c * 128;
      float xy = fminf(mdL[wid][e] + 1.f, -EPSF) - 1.f;
      float u[4], p = 0.f;
#pragma unroll
      for (int j = 0; j < 4; j++) {
        int cc = lane * 4 + j;
        u[j] = xc[cc] + xy * xr[j];
        p += (cc == 0) ? -u[j] * u[j] : u[j] * u[j];  // mink_dot(u,u)
      }
      float mn = wsum32(p);
      float nu = fmaxf(sqrtf(fmaxf(mn, MIN_NORM)), MIN_NORM);
      float scale = sqrtf(geoL[wid][e]) / nu * attL[wid][e];
#pragma unroll
      for (int j = 0; j < 4; j++) aggv[j] += scale * u[j];
    }
    __syncthreads();
  }
#pragma unroll
  for (int j = 0; j < 4; j++)
    out_agg[(size_t)r * 128 + lane * 4 + j] = aggv[j] * (1.f / 100.f);
}

// =====================================================================
// Kernel 3: per-node tail: x=expmap(proj_tan(agg,x),x); layernorm(logmap0);
// x=expmap0(...); x=expmap0(proj_tan0(silu(logmap0(x)))).
// =====================================================================
__global__ void node_update_kernel(const float* x_in, const float* agg,
                                   const float* __restrict__ g,
                                   const float* __restrict__ be,
                                   float* x_out) {
  const int lane = threadIdx.x & 31;
  const int wid  = threadIdx.x >> 5;
  const int node = blockIdx.x * 4 + wid;
  const float* xr = x_in + (size_t)node * 128;
  const float* ar = agg  + (size_t)node * 128;
  float xv[4], ov[4];
#pragma unroll
  for (int j = 0; j < 4; j++) { xv[j] = xr[lane * 4 + j]; ov[j] = ar[lane * 4 + j]; }
  float x0 = __shfl(xv[0], 0, 32);
  // proj_tan(agg, x)
  float part = 0.f;
#pragma unroll
  for (int j = 0; j < 4; j++) { int c = lane * 4 + j; if (c > 0) part += xv[j] * ov[j]; }
  float t0 = wsum32(part) / fmaxf(x0, EPSF);
  // expmap(t, x)
  part = 0.f;
#pragma unroll
  for (int j = 0; j < 4; j++) { int c = lane * 4 + j; if (c > 0) part += ov[j] * ov[j]; }
  float md = wsum32(part) - t0 * t0;
  float nu = fminf(sqrtf(fmaxf(md, MIN_NORM)), MAX_NORMF);
  float theta = fmaxf(nu, MIN_NORM);
  float ch = coshf(theta), shk = sinhf(theta) / theta;
#pragma unroll
  for (int j = 0; j < 4; j++) { int c = lane * 4 + j; if (c > 0) xv[j] = ch * xv[j] + shk * ov[j]; }
  part = 0.f;
#pragma unroll
  for (int j = 0; j < 4; j++) { int c = lane * 4 + j; if (c > 0) part += xv[j] * xv[j]; }
  float s2  = wsum32(part);
  float nx0 = sqrtf(1.f + s2);
  // ht = logmap0(x)
  float yn = fmaxf(sqrtf(s2), MIN_NORM);
  float sc = arcosh_f(fmaxf(nx0, 1.f + EPSF)) / yn;
  float ht[4];
#pragma unroll
  for (int j = 0; j < 4; j++) { int c = lane * 4 + j; ht[j] = (c == 0) ? 0.f : sc * xv[j]; }
  // layernorm over components 1..127
  float sA = 0.f, sB = 0.f;
#pragma unroll
  for (int j = 0; j < 4; j++) { int c = lane * 4 + j; if (c > 0) { sA += ht[j]; sB += ht[j] * ht[j]; } }
  float S = wsum32(sA), S2 = wsum32(sB);
  float mu = S / 127.f;
  float var = S2 / 127.f - mu * mu;
  float inv = rsqrtf(var + LN_EPSF);
#pragma unroll
  for (int j = 0; j < 4; j++) {
    int c = lane * 4 + j;
    if (c > 0) ht[j] = (ht[j] - mu) * inv * g[c - 1] + be[c - 1];
  }
  // x = expmap0(ht)
  part = 0.f;
#pragma unroll
  for (int j = 0; j < 4; j++) { int c = lane * 4 + j; if (c > 0) part += ht[j] * ht[j]; }
  float xn = fmaxf(sqrtf(wsum32(part)), MIN_NORM);
  float ch2 = coshf(xn), sh2 = sinhf(xn) / xn;
#pragma unroll
  for (int j = 0; j < 4; j++) { int c = lane * 4 + j; xv[j] = (c == 0) ? ch2 : sh2 * ht[j]; }
  // x = expmap0(proj_tan0(silu(logmap0(x))))
  part = 0.f;
#pragma unroll
  for (int j = 0; j < 4; j++) { int c = lane * 4 + j; if (c > 0) part += xv[j] * xv[j]; }
  float ynb = fmaxf(sqrtf(wsum32(part)), MIN_NORM);
  float sc3 = arcosh_f(fmaxf(ch2, 1.f + EPSF)) / ynb;
  float sl[4];
#pragma unroll
  for (int j = 0; j < 4; j++) { int c = lane * 4 + j; sl[j] = (c == 0) ? 0.f : siluf(sc3 * xv[j]); }
  part = 0.f;
#pragma unroll
  for (int j = 0; j < 4; j++) { int c = lane * 4 + j; if (c > 0) part += sl[j] * sl[j]; }
  float xnf = fmaxf(sqrtf(wsum32(part)), MIN_NORM);
  float chf = coshf(xnf), shf = sinhf(xnf) / xnf;
  float* xo = x_out + (size_t)node * 128;
#pragma unroll
  for (int j = 0; j < 4; j++) { int c = lane * 4 + j; xo[c] = (c == 0) ? chf : shf * sl[j]; }
}

// =====================================================================
extern "C" void kernel_launch(void* const* d_in, const int* in_sizes, int n_in,
                              void* d_out, int out_size, void* d_ws, size_t ws_size,
                              hipStream_t stream) {
  (void)in_sizes; (void)n_in; (void)out_size; (void)ws_size;
  const float* h      = (const float*)d_in[0];
  const float* dists  = (const float*)d_in[1];
  /* d_in[2] node_mask unused by reference */
  const float* emask  = (const float*)d_in[3];
  const float* W0     = (const float*)d_in[4];
  const float* b0     = (const float*)d_in[5];
  const float* aW1_0  = (const float*)d_in[6];
  const float* ab1_0  = (const float*)d_in[7];
  const float* aW2_0  = (const float*)d_in[8];
  const float* ab2_0  = (const float*)d_in[9];
  const float* g0     = (const float*)d_in[10];
  const float* be0    = (const float*)d_in[11];
  const float* Ws     = (const float*)d_in[12];
  const float* bs     = (const float*)d_in[13];
  const float* aW1s   = (const float*)d_in[14];
  const float* ab1s   = (const float*)d_in[15];
  const float* aW2s   = (const float*)d_in[16];
  const float* ab2s   = (const float*)d_in[17];
  const float* gs     = (const float*)d_in[18];
  const float* bes    = (const float*)d_in[19];
  const int*   row    = (const int*)d_in[20];
  const int*   col    = (const int*)d_in[21];
  float* out = (float*)d_out;

  const int N = 4096, Dm = 128;
  float* ws    = (float*)d_ws;
  float* x     = ws;                   // N*128
  float* xt    = x  + (size_t)N * Dm;  // N*128
  float* agg   = xt + (size_t)N * Dm;  // N*128
  float* Bpack = agg + (size_t)N * Dm; // 8*NKG*32*2 = KPAD*128 floats

  for (int layer = 0; layer < 5; ++layer) {
    const float *W, *bias, *aW1, *ab1, *aW2, *ab2, *g, *be;
    int Din;
    const float* hin;
    if (layer == 0) {
      Din = 20; hin = h;
      W = W0; bias = b0; aW1 = aW1_0; ab1 = ab1_0; aW2 = aW2_0; ab2 = ab2_0;
      g = g0; be = be0;
    } else {
      int i = layer - 1;
      Din = 128; hin = x;
      W   = Ws   + (size_t)i * 128 * 128;
      bias= bs   + (size_t)i * 128;
      aW1 = aW1s + (size_t)i * 128 * 258;
      ab1 = ab1s + (size_t)i * 128;
      aW2 = aW2s + (size_t)i * 128;
      ab2 = ab2s + (size_t)i;
      g   = gs   + (size_t)i * 127;
      be  = bes  + (size_t)i * 127;
    }
    pack_aW1_kernel<<<(8 * NKG * 32 + 255) / 256, 256, 0, stream>>>(aW1, Bpack);
    node_linear_kernel<<<N / 4, 128, 0, stream>>>(hin, Din, W, bias, x, xt);
    edge_attn_kernel<<<N / 2, 64, 0, stream>>>(x, xt, dists, emask, Bpack,
                                               ab1, aW2, ab2, row, col, agg);
    node_update_kernel<<<N / 4, 128, 0, stream>>>(x, agg, g, be,
                                                  (layer == 4) ? out : x);
  }
}